// LocalModel_74947179316231
// MI455X (gfx1250) — compile-verified
//
#include <hip/hip_runtime.h>

#define D 128
#define BN_EPS 1e-5f
#define WAVES 8          // waves per block in the GEMM kernel (8*16 = 128 rows/block)

typedef __attribute__((ext_vector_type(2))) float v2f;
typedef __attribute__((ext_vector_type(4))) float v4f;
typedef __attribute__((ext_vector_type(8))) float v8f;

// ---------------------------------------------------------------------------
// Kernel 1: zero the aggregation buffer (d_out doubles as agg) and BN stats
// ---------------------------------------------------------------------------
__global__ void zero_kernel(float* __restrict__ out, float* __restrict__ stats, int n_out) {
    int idx = blockIdx.x * blockDim.x + threadIdx.x;
    int stride = gridDim.x * blockDim.x;
    for (int i = idx; i < n_out; i += stride) out[i] = 0.0f;
    if (idx < 2 * D) stats[idx] = 0.0f;
}

// ---------------------------------------------------------------------------
// Kernel 2: edge scatter. One wave32 per edge; lane l owns cols 4l..4l+3.
// msg = relu(x[src] + edge_attr); atomicAdd into agg[dst] (L2-resident).
// edge_attr is streamed once -> non-temporal loads keep x/agg hot in L2.
// ---------------------------------------------------------------------------
__global__ void edge_scatter_kernel(const float* __restrict__ x,
                                    const int* __restrict__ src,
                                    const int* __restrict__ dst,
                                    const float* __restrict__ ea,
                                    float* __restrict__ agg, int E_) {
    int wave = (blockIdx.x * blockDim.x + threadIdx.x) >> 5;
    int lane = threadIdx.x & 31;
    if (wave >= E_) return;
    int s = src[wave];
    int d = dst[wave];
    int c = lane * 4;
    const v4f xv = *(const v4f*)(x + (size_t)s * D + c);
    const v4f* eap = (const v4f*)(ea + (size_t)wave * D + c);
    v4f av = __builtin_nontemporal_load(eap);
    v4f m;
    m.x = fmaxf(xv.x + av.x, 0.0f);
    m.y = fmaxf(xv.y + av.y, 0.0f);
    m.z = fmaxf(xv.z + av.z, 0.0f);
    m.w = fmaxf(xv.w + av.w, 0.0f);
    float* p = agg + (size_t)d * D + c;
    atomicAdd(p + 0, m.x);
    atomicAdd(p + 1, m.y);
    atomicAdd(p + 2, m.z);
    atomicAdd(p + 3, m.w);
}

// ---------------------------------------------------------------------------
// Kernel 3: fused  h0 = x + agg;  hid = relu(h0@W1+b1);  out = hid@W2+b2+x
// Per wave: one 16-row tile, full K=128 via 32 chained V_WMMA_F32_16X16X4_F32
// per 16-col output tile (8 col tiles held in v8f accumulators).
// agg lives in `out`; each wave reads only its own rows before overwriting.
//
// Fragment layouts per CDNA5 ISA 7.12.2 (f32, wave32):
//   A 16x4 : lane(0-15)=M, lane(16-31)=M w/ K+2 ; vgpr v -> K = 4*kt + 2*hi + v
//   B 4x16 : lane&15 = N, hi selects K+2      ; vgpr v -> K = 4*kt + 2*hi + v
//   C 16x16: vgpr r, lane&15 = N, M = r + 8*hi
// ---------------------------------------------------------------------------
__global__ __launch_bounds__(WAVES * 32, 1)
void gine_mlp_kernel(const float* __restrict__ x,
                     const float* __restrict__ W1, const float* __restrict__ b1,
                     const float* __restrict__ W2, const float* __restrict__ b2,
                     float* __restrict__ out, int n) {
    extern __shared__ float smem[];
    float* ws   = smem;            // 128x128 weight stage (64KB), reused W1 then W2
    float* tile = smem + D * D;    // WAVES x (16 x 128) A/hidden tiles (64KB)

    const int t    = threadIdx.x;
    const int wave = t >> 5;
    const int lane = t & 31;
    const int hi   = lane >> 4;    // 0: lanes 0-15, 1: lanes 16-31
    const int ln   = lane & 15;
    const int r0   = (blockIdx.x * WAVES + wave) * 16;
    const bool act = (r0 < n);
    float* ta = tile + wave * (16 * D);

    // ---- stage W1 into LDS (all threads) ----
    for (int i = t; i < (D * D) / 4; i += WAVES * 32)
        ((v4f*)ws)[i] = ((const v4f*)W1)[i];
    __syncthreads();

    v8f acc[8];
    if (act) {
        // stage A tile: h0 = x + agg  (agg is in `out`)
        for (int i = lane; i < 16 * (D / 4); i += 32) {
            int row = i >> 5;           // 0..15
            int c   = (i & 31) * 4;     // 0..124
            const v4f xv = *(const v4f*)(x   + (size_t)(r0 + row) * D + c);
            const v4f gv = *(const v4f*)(out + (size_t)(r0 + row) * D + c);
            *(v4f*)(ta + row * D + c) = xv + gv;
        }

        v8f z = {};
#pragma unroll
        for (int nt = 0; nt < 8; ++nt) acc[nt] = z;

        // GEMM1: hid_tile = A(16x128) @ W1(128x128)
        for (int kt = 0; kt < 32; ++kt) {
            const int kb = kt * 4 + 2 * hi;
            v2f a;
            a.x = ta[ln * D + kb];
            a.y = ta[ln * D + kb + 1];
#pragma unroll
            for (int nt = 0; nt < 8; ++nt) {
                const int col = nt * 16 + ln;
                v2f b;
                b.x = ws[kb * D + col];
                b.y = ws[(kb + 1) * D + col];
                acc[nt] = __builtin_amdgcn_wmma_f32_16x16x4_f32(
                    false, a, false, b, (short)0, acc[nt], false, false);
            }
        }

        // bias + ReLU, write hidden tile back over A tile (same wave, in-order DS)
#pragma unroll
        for (int nt = 0; nt < 8; ++nt) {
            const int col = nt * 16 + ln;
            const float bias = b1[col];
#pragma unroll
            for (int r = 0; r < 8; ++r) {
                const int m = r + 8 * hi;
                ta[m * D + col] = fmaxf(acc[nt][r] + bias, 0.0f);
            }
        }
    }
    __syncthreads();

    // ---- restage: W2 into LDS (all threads) ----
    for (int i = t; i < (D * D) / 4; i += WAVES * 32)
        ((v4f*)ws)[i] = ((const v4f*)W2)[i];
    __syncthreads();

    if (act) {
        v8f acc2[8];
        v8f z = {};
#pragma unroll
        for (int nt = 0; nt < 8; ++nt) acc2[nt] = z;

        // GEMM2: out_tile = hid(16x128) @ W2(128x128)
        for (int kt = 0; kt < 32; ++kt) {
            const int kb = kt * 4 + 2 * hi;
            v2f a;
            a.x = ta[ln * D + kb];
            a.y = ta[ln * D + kb + 1];
#pragma unroll
            for (int nt = 0; nt < 8; ++nt) {
                const int col = nt * 16 + ln;
                v2f b;
                b.x = ws[kb * D + col];
                b.y = ws[(kb + 1) * D + col];
                acc2[nt] = __builtin_amdgcn_wmma_f32_16x16x4_f32(
                    false, a, false, b, (short)0, acc2[nt], false, false);
            }
        }

        // epilogue: + b2 + residual x, store pre-BN result
#pragma unroll
        for (int nt = 0; nt < 8; ++nt) {
            const int col = nt * 16 + ln;
            const float bias = b2[col];
#pragma unroll
            for (int r = 0; r < 8; ++r) {
                const int m   = r + 8 * hi;
                const size_t o = (size_t)(r0 + m) * D + col;
                out[o] = acc2[nt][r] + bias + x[o];
            }
        }
    }
}

// ---------------------------------------------------------------------------
// Kernel 4: per-column sum / sumsq (128 threads = 1 col each, row-strided)
// ---------------------------------------------------------------------------
__global__ void stats_kernel(const float* __restrict__ h, float* __restrict__ stats, int n) {
    const int col = threadIdx.x;   // 0..127
    float s = 0.0f, s2 = 0.0f;
    for (int row = blockIdx.x; row < n; row += gridDim.x) {
        const float v = h[(size_t)row * D + col];
        s  += v;
        s2 += v * v;
    }
    atomicAdd(&stats[col], s);
    atomicAdd(&stats[D + col], s2);
}

// ---------------------------------------------------------------------------
// Kernel 5: BatchNorm normalize in place (training-mode batch stats)
// ---------------------------------------------------------------------------
__global__ void bn_kernel(float* __restrict__ h, const float* __restrict__ stats,
                          const float* __restrict__ gamma, const float* __restrict__ beta,
                          int n) {
    const int col = threadIdx.x;   // 0..127
    const float invN = 1.0f / (float)n;
    const float mean = stats[col] * invN;
    const float var  = stats[D + col] * invN - mean * mean;
    const float g    = gamma[col] * rsqrtf(var + BN_EPS);
    const float bta  = beta[col];
    for (int row = blockIdx.x; row < n; row += gridDim.x) {
        const size_t i = (size_t)row * D + col;
        h[i] = (h[i] - mean) * g + bta;
    }
}

// ---------------------------------------------------------------------------
extern "C" void kernel_launch(void* const* d_in, const int* in_sizes, int n_in,
                              void* d_out, int out_size, void* d_ws, size_t ws_size,
                              hipStream_t stream) {
    const float* x     = (const float*)d_in[0];
    const int*   ei    = (const int*)d_in[1];   // [2, E] -> src row, dst row
    const float* ea    = (const float*)d_in[2];
    const float* W1    = (const float*)d_in[3];
    const float* b1    = (const float*)d_in[4];
    const float* W2    = (const float*)d_in[5];
    const float* b2    = (const float*)d_in[6];
    const float* gamma = (const float*)d_in[7];
    const float* beta  = (const float*)d_in[8];

    float* out   = (float*)d_out;
    float* stats = (float*)d_ws;    // [2*D] floats

    const int n = in_sizes[0] / D;  // 50000
    const int e = in_sizes[2] / D;  // 800000
    const int* src = ei;
    const int* dst = ei + e;

    // 1) zero agg (in d_out) + stats
    zero_kernel<<<2048, 256, 0, stream>>>(out, stats, n * D);

    // 2) edge scatter: 8 waves/block, 1 wave per edge
    edge_scatter_kernel<<<(e + 7) / 8, 256, 0, stream>>>(x, src, dst, ea, out, e);

    // 3) fused GEMM/MLP/residual (128 rows per block; N is a multiple of 16)
    const int gemm_blocks = (n + WAVES * 16 - 1) / (WAVES * 16);
    const size_t lds_bytes = (size_t)(D * D + WAVES * 16 * D) * sizeof(float); // 128 KB
    gine_mlp_kernel<<<gemm_blocks, WAVES * 32, lds_bytes, stream>>>(
        x, W1, b1, W2, b2, out, n);

    // 4) batch statistics
    stats_kernel<<<512, D, 0, stream>>>(out, stats, n);

    // 5) normalize
    bn_kernel<<<1024, D, 0, stream>>>(out, stats, gamma, beta, n);
}